// DistributedFNONd_80960133529921
// MI455X (gfx1250) — compile-verified
//
#include <hip/hip_runtime.h>

// ---------------------------------------------------------------------------
// DistributedFNONd forward for MI455X (gfx1250), fp32 end-to-end.
// Matrix ops use V_WMMA_F32_16X16X4_F32 (full fp32 WMMA, wave32).
// ---------------------------------------------------------------------------

#define PI_F 3.14159265358979323846f

// Problem constants (match reference)
#define NB    8        // batch
#define CH    32       // width
#define NXD   64
#define NYD   64
#define TI    10
#define TO    20
#define MXM   12
#define MYM   12
#define MTM   8
#define KXM   24       // kept X modes (low 12 + high 12)
#define KYM   24
#define FCH   128
#define NP0   (NXD*NYD*TI)   // 40960
#define NPT   (NXD*NYD*TO)   // 81920
#define MODEN (KXM*KYM*MTM)  // 4608
#define MSZ   (MXM*MYM*MTM)  // 1152

typedef __attribute__((ext_vector_type(2))) float v2f;
typedef __attribute__((ext_vector_type(8))) float v8f;

static __device__ __forceinline__ v8f wmma_f32(v2f a, v2f b, v8f c) {
  // D = A(16x4,f32) * B(4x16,f32) + C(16x16,f32)
  return __builtin_amdgcn_wmma_f32_16x16x4_f32(false, a, false, b, (short)0, c,
                                               false, false);
}

static __device__ __forceinline__ float gelu_f(float x) {
  return 0.5f * x * (1.0f + erff(x * 0.70710678118654752440f));
}

// ---------------------------------------------------------------------------
// Repack spectral weights [blk,corner,ci,co,mx,my,kt] -> [blk,corner,mode,ci,co]
// so per-mode WMMA B-fragment loads are lane-contiguous.
// ---------------------------------------------------------------------------
__global__ void k_repack(const float* __restrict__ sre,
                         const float* __restrict__ sim,
                         float* __restrict__ wre, float* __restrict__ wim) {
  int i = blockIdx.x * 256 + threadIdx.x;   // 18,874,368 total (exact grid)
  int kt = i % 8;       int r = i / 8;
  int my = r % 12;      r /= 12;
  int mx = r % 12;      r /= 12;
  int co = r % 32;      r /= 32;
  int ci = r % 32;      r /= 32;
  int corner = r % 4;   int blk = r / 4;
  int dst = ((blk * 4 + corner) * MSZ + (mx * 12 + my) * 8 + kt) * 1024 +
            ci * 32 + co;
  wre[dst] = sre[i];
  wim[dst] = sim[i];
}

// ---------------------------------------------------------------------------
// Channel-axis GEMM with f32 WMMA (compile-time shapes -> fully unrolled):
//   out[b,co,p] = (gelu?)( sum_ci in[b,ci,p]*W[ci,co] + bias[co] )
// Layout: [B, C, NPl]. 256 threads = 8 waves; wave handles 16 points x Cout.
// ---------------------------------------------------------------------------
template <int Cin, int Cout, int DoGelu>
__global__ void k_chgemm(const float* __restrict__ in,
                         const float* __restrict__ Wm,
                         const float* __restrict__ bias,
                         float* __restrict__ out, int NPl) {
  __shared__ float sW[Cin * Cout];
  __shared__ float sB[Cout];
  for (int i = threadIdx.x; i < Cin * Cout; i += blockDim.x) sW[i] = Wm[i];
  for (int i = threadIdx.x; i < Cout; i += blockDim.x) sB[i] = bias[i];
  __syncthreads();

  int wave = threadIdx.x >> 5, lane = threadIdx.x & 31;
  int tilesPerB = NPl >> 4;
  int tile = blockIdx.x * 8 + wave;                 // grid sized exactly
  int b = tile / tilesPerB;
  int p0 = (tile - b * tilesPerB) << 4;
  const float* inb = in + (long)b * Cin * NPl;
  float* outb = out + (long)b * Cout * NPl;
  int M = lane & 15;
  int khalf = (lane < 16) ? 0 : 2;

  // Preload all A fragments once (reused across N-tiles)
  v2f av[Cin / 4];
#pragma unroll
  for (int ks = 0; ks < Cin / 4; ++ks) {
    int k = ks * 4 + khalf;
    av[ks][0] = inb[(long)k * NPl + p0 + M];
    av[ks][1] = inb[(long)(k + 1) * NPl + p0 + M];
    // stream next tile of this channel row into cache (global_prefetch_b8)
    __builtin_prefetch(&inb[(long)k * NPl + p0 + M + 128], 0, 1);
  }

#pragma unroll
  for (int t = 0; t < Cout / 16; ++t) {
    int co0 = t << 4;
    v8f acc = {};
#pragma unroll
    for (int ks = 0; ks < Cin / 4; ++ks) {
      int k = ks * 4 + khalf;
      v2f bb; bb[0] = sW[k * Cout + co0 + M];
              bb[1] = sW[(k + 1) * Cout + co0 + M];
      acc = wmma_f32(av[ks], bb, acc);
    }
    float bv = sB[co0 + M];
    float* op = outb + (long)(co0 + M) * NPl + p0 + ((lane < 16) ? 0 : 8);
#pragma unroll
    for (int r = 0; r < 8; ++r) {
      float v = acc[r] + bv;
      if (DoGelu) v = gelu_f(v);
      op[r] = v;
    }
  }
}

// ---------------------------------------------------------------------------
// Time-lift GEMM (contract last axis): out[q,to]=gelu(sum_ti in[q,ti]W[ti,to]+b)
// q = (b,c,x,y), NQ = 1,048,576. K padded 10->12, N padded 20->32.
// ---------------------------------------------------------------------------
__global__ void k_timelift(const float* __restrict__ in,
                           const float* __restrict__ W,
                           const float* __restrict__ bias,
                           float* __restrict__ out) {
  __shared__ float sW[12 * 32];
  __shared__ float sB[32];
  for (int i = threadIdx.x; i < 12 * 32; i += blockDim.x) {
    int k = i >> 5, n = i & 31;
    sW[i] = (k < TI && n < TO) ? W[k * TO + n] : 0.0f;
  }
  for (int i = threadIdx.x; i < 32; i += blockDim.x)
    sB[i] = (i < TO) ? bias[i] : 0.0f;
  __syncthreads();

  int wave = threadIdx.x >> 5, lane = threadIdx.x & 31;
  int tile = blockIdx.x * 8 + wave;   // 65536 tiles exact
  long q0 = (long)tile << 4;
  int M = lane & 15;
  int khalf = (lane < 16) ? 0 : 2;

  // Preload A fragments once (K = 12 padded, 3 steps)
  v2f av[3];
#pragma unroll
  for (int ks = 0; ks < 3; ++ks) {
    int k = ks * 4 + khalf;
    av[ks][0] = (k < TI) ? in[(q0 + M) * TI + k] : 0.0f;
    av[ks][1] = (k + 1 < TI) ? in[(q0 + M) * TI + k + 1] : 0.0f;
  }
  __builtin_prefetch(&in[(q0 + 128) * TI], 0, 1);

#pragma unroll
  for (int t = 0; t < 2; ++t) {
    int co0 = t << 4;
    v8f acc = {};
#pragma unroll
    for (int ks = 0; ks < 3; ++ks) {
      int k = ks * 4 + khalf;
      v2f bb; bb[0] = sW[k * 32 + co0 + M];
              bb[1] = sW[(k + 1) * 32 + co0 + M];
      acc = wmma_f32(av[ks], bb, acc);
    }
    int col = co0 + M;
    if (col < TO) {
      float bv = sB[col];
      long qb = q0 + ((lane < 16) ? 0 : 8);
#pragma unroll
      for (int r = 0; r < 8; ++r)
        out[(qb + r) * TO + col] = gelu_f(acc[r] + bv);
    }
  }
}

// ---------------------------------------------------------------------------
// Forward T-DFT (rfft, keep kt=0..7): x[bc,x,y,t] -> Ft[bc,y,kt,x] (re/im)
// ---------------------------------------------------------------------------
__global__ void k_tfwd(const float* __restrict__ x,
                       float* __restrict__ ftre, float* __restrict__ ftim) {
  __shared__ float ct[MTM * TO], st[MTM * TO];
  if (threadIdx.x < MTM * TO) {
    int k = threadIdx.x / TO, t = threadIdx.x % TO;
    float ang = -2.0f * PI_F * (float)(k * t) / (float)TO;
    float s, c; sincosf(ang, &s, &c);
    ct[threadIdx.x] = c; st[threadIdx.x] = s;
  }
  __syncthreads();
  int g = blockIdx.x * 256 + threadIdx.x;   // 1,048,576 exact
  int xg = g & 63, y = (g >> 6) & 63, bc = g >> 12;
  const float* src = x + (((long)(bc * 64 + xg) * 64 + y) * TO);
  float fr[MTM], fi[MTM];
#pragma unroll
  for (int k = 0; k < MTM; ++k) { fr[k] = 0.0f; fi[k] = 0.0f; }
  for (int t = 0; t < TO; ++t) {
    float v = src[t];
#pragma unroll
    for (int k = 0; k < MTM; ++k) {
      fr[k] += v * ct[k * TO + t];
      fi[k] += v * st[k * TO + t];
    }
  }
  long ob = ((long)(bc * 64 + y) * MTM) * 64 + xg;
#pragma unroll
  for (int k = 0; k < MTM; ++k) {
    ftre[ob + k * 64] = fr[k];
    ftim[ob + k * 64] = fi[k];
  }
}

// ---------------------------------------------------------------------------
// Forward X-DFT (24 kept modes): Ft[bc,y,kt,x] -> Fx[bc,kxi,kt,y]
// One block per (bc,y); 192 threads = (kxi,kt).
// ---------------------------------------------------------------------------
__global__ void k_xfwd(const float* __restrict__ ftre,
                       const float* __restrict__ ftim,
                       float* __restrict__ fxre, float* __restrict__ fxim) {
  __shared__ float lr[512], li[512], c64[64], s64[64];
  int bc = blockIdx.x >> 6, y = blockIdx.x & 63;
  if (threadIdx.x < 64) {
    float ang = 2.0f * PI_F * (float)threadIdx.x / 64.0f;
    float s, c; sincosf(ang, &s, &c);
    c64[threadIdx.x] = c; s64[threadIdx.x] = s;
  }
  long base = ((long)(bc * 64 + y) * MTM) * 64;
  for (int i = threadIdx.x; i < 512; i += 192) {
    lr[i] = ftre[base + i]; li[i] = ftim[base + i];
  }
  __syncthreads();
  int kxi = threadIdx.x / 8, kt = threadIdx.x & 7;
  int kx = (kxi < 12) ? kxi : kxi + 40;
  float re = 0.0f, im = 0.0f;
  for (int xg = 0; xg < 64; ++xg) {
    int j = (kx * xg) & 63;
    float fr = lr[kt * 64 + xg], fi = li[kt * 64 + xg];
    re += fr * c64[j] + fi * s64[j];        // e^{-i theta}
    im += fi * c64[j] - fr * s64[j];
  }
  long o = ((long)(bc * KXM + kxi) * MTM + kt) * 64 + y;
  fxre[o] = re; fxim[o] = im;
}

// ---------------------------------------------------------------------------
// Forward Y-DFT: Fx[bc,kxi,kt,y] -> Fxy[mode=(kxi,kyi,kt), ci, b]
// One block per (bc,kxi); 192 threads = (kyi,kt).
// ---------------------------------------------------------------------------
__global__ void k_yfwd(const float* __restrict__ fxre,
                       const float* __restrict__ fxim,
                       float* __restrict__ fyre, float* __restrict__ fyim) {
  __shared__ float lr[512], li[512], c64[64], s64[64];
  int bc = blockIdx.x / KXM, kxi = blockIdx.x % KXM;
  if (threadIdx.x < 64) {
    float ang = 2.0f * PI_F * (float)threadIdx.x / 64.0f;
    float s, c; sincosf(ang, &s, &c);
    c64[threadIdx.x] = c; s64[threadIdx.x] = s;
  }
  long base = ((long)(bc * KXM + kxi) * MTM) * 64;
  for (int i = threadIdx.x; i < 512; i += 192) {
    lr[i] = fxre[base + i]; li[i] = fxim[base + i];
  }
  __syncthreads();
  int kyi = threadIdx.x / 8, kt = threadIdx.x & 7;
  int ky = (kyi < 12) ? kyi : kyi + 40;
  float re = 0.0f, im = 0.0f;
  for (int y = 0; y < 64; ++y) {
    int j = (ky * y) & 63;
    float fr = lr[kt * 64 + y], fi = li[kt * 64 + y];
    re += fr * c64[j] + fi * s64[j];
    im += fi * c64[j] - fr * s64[j];
  }
  int b = bc >> 5, c = bc & 31;
  long o = ((long)((kxi * KYM + kyi) * MTM + kt) * CH + c) * NB + b;
  fyre[o] = re; fyim[o] = im;
}

// ---------------------------------------------------------------------------
// Per-mode complex channel mix on WMMA (complex packed into M):
//   A' = [Re(8x32); Im(8x32)] (16x32), D1 = A'*Wre, D2 = A'*Wim
//   out_re = D1[0:8] - D2[8:16], out_im = D2[0:8] + D1[8:16]  (shfl_xor 16)
// One wave per mode; output G[b,co,kxi,kyi,kt].
// ---------------------------------------------------------------------------
__global__ void k_mix(const float* __restrict__ fre,
                      const float* __restrict__ fim,
                      const float* __restrict__ wre,
                      const float* __restrict__ wim,
                      float* __restrict__ gre, float* __restrict__ gim,
                      int blk) {
  int wave = threadIdx.x >> 5, lane = threadIdx.x & 31;
  int m = blockIdx.x * 8 + wave;              // 4608 modes exact
  int kxi = m / (KYM * MTM);
  int rem = m - kxi * (KYM * MTM);
  int kyi = rem >> 3, kt = rem & 7;
  int corner = ((kxi >= 12) ? 2 : 0) + ((kyi >= 12) ? 1 : 0);
  int mx = (kxi >= 12) ? kxi - 12 : kxi;
  int my = (kyi >= 12) ? kyi - 12 : kyi;
  long wb = ((long)(blk * 4 + corner) * MSZ + (mx * 12 + my) * 8 + kt) * 1024;
  long fb = (long)m * (CH * NB);
  int M = lane & 15;
  int khalf = (lane < 16) ? 0 : 2;
  const float* ap = ((M < 8) ? fre : fim) + fb + (M & 7);
  const float* wrp = wre + wb;
  const float* wip = wim + wb;

  // Preload all A fragments (8 K-steps), then run 32 back-to-back WMMAs.
  v2f av[8];
#pragma unroll
  for (int ks = 0; ks < 8; ++ks) {
    int k = ks * 4 + khalf;
    av[ks][0] = ap[k * NB];
    av[ks][1] = ap[(k + 1) * NB];
  }

  v8f d1a = {}, d1b = {}, d2a = {}, d2b = {};
#pragma unroll
  for (int ks = 0; ks < 8; ++ks) {
    int k = ks * 4 + khalf;
    v2f br0; br0[0] = wrp[k * 32 + M];      br0[1] = wrp[(k + 1) * 32 + M];
    v2f br1; br1[0] = wrp[k * 32 + 16 + M]; br1[1] = wrp[(k + 1) * 32 + 16 + M];
    v2f bi0; bi0[0] = wip[k * 32 + M];      bi0[1] = wip[(k + 1) * 32 + M];
    v2f bi1; bi1[0] = wip[k * 32 + 16 + M]; bi1[1] = wip[(k + 1) * 32 + 16 + M];
    d1a = wmma_f32(av[ks], br0, d1a);
    d1b = wmma_f32(av[ks], br1, d1b);
    d2a = wmma_f32(av[ks], bi0, d2a);
    d2b = wmma_f32(av[ks], bi1, d2b);
  }
#pragma unroll
  for (int r = 0; r < 8; ++r) {
    float s1a = __shfl_xor(d1a[r], 16, 32);
    float s2a = __shfl_xor(d2a[r], 16, 32);
    float s1b = __shfl_xor(d1b[r], 16, 32);
    float s2b = __shfl_xor(d2b[r], 16, 32);
    if (lane < 16) {
      float re0 = d1a[r] - s2a, im0 = d2a[r] + s1a;
      float re1 = d1b[r] - s2b, im1 = d2b[r] + s1b;
      long i0 = (((long)(r * CH + lane) * KXM + kxi) * KYM + kyi) * MTM + kt;
      long i1 = (((long)(r * CH + 16 + lane) * KXM + kxi) * KYM + kyi) * MTM + kt;
      gre[i0] = re0; gim[i0] = im0;
      gre[i1] = re1; gim[i1] = im1;
    }
  }
}

// ---------------------------------------------------------------------------
// Inverse Y-DFT: G[bc,kxi,kyi,kt] -> Gy[bc,y,kxi,kt]   (e^{+i theta})
// One block per (bc,kxi); 512 threads = (y,kt).
// ---------------------------------------------------------------------------
__global__ void k_yinv(const float* __restrict__ gre,
                       const float* __restrict__ gim,
                       float* __restrict__ gyre, float* __restrict__ gyim) {
  __shared__ float lr[KYM * MTM], li[KYM * MTM], c64[64], s64[64];
  int bc = blockIdx.x / KXM, kxi = blockIdx.x % KXM;
  if (threadIdx.x < 64) {
    float ang = 2.0f * PI_F * (float)threadIdx.x / 64.0f;
    float s, c; sincosf(ang, &s, &c);
    c64[threadIdx.x] = c; s64[threadIdx.x] = s;
  }
  long base = (long)(bc * KXM + kxi) * (KYM * MTM);
  if (threadIdx.x < KYM * MTM) {
    lr[threadIdx.x] = gre[base + threadIdx.x];
    li[threadIdx.x] = gim[base + threadIdx.x];
  }
  __syncthreads();
  int y = threadIdx.x >> 3, kt = threadIdx.x & 7;
  float re = 0.0f, im = 0.0f;
  for (int kyi = 0; kyi < KYM; ++kyi) {
    int ky = (kyi < 12) ? kyi : kyi + 40;
    int j = (ky * y) & 63;
    float gr = lr[kyi * 8 + kt], gi = li[kyi * 8 + kt];
    re += gr * c64[j] - gi * s64[j];
    im += gr * s64[j] + gi * c64[j];
  }
  long o = ((long)(bc * 64 + y) * KXM + kxi) * MTM + kt;
  gyre[o] = re; gyim[o] = im;
}

// ---------------------------------------------------------------------------
// Inverse X-DFT: Gy[bc,y,kxi,kt] -> G2[bc,x,y,kt]   (e^{+i theta})
// One block per (bc,y); 512 threads = (x,kt).
// ---------------------------------------------------------------------------
__global__ void k_xinv(const float* __restrict__ gyre,
                       const float* __restrict__ gyim,
                       float* __restrict__ g2re, float* __restrict__ g2im) {
  __shared__ float lr[KXM * MTM], li[KXM * MTM], c64[64], s64[64];
  int bc = blockIdx.x >> 6, y = blockIdx.x & 63;
  if (threadIdx.x < 64) {
    float ang = 2.0f * PI_F * (float)threadIdx.x / 64.0f;
    float s, c; sincosf(ang, &s, &c);
    c64[threadIdx.x] = c; s64[threadIdx.x] = s;
  }
  long base = (long)(bc * 64 + y) * (KXM * MTM);
  if (threadIdx.x < KXM * MTM) {
    lr[threadIdx.x] = gyre[base + threadIdx.x];
    li[threadIdx.x] = gyim[base + threadIdx.x];
  }
  __syncthreads();
  int xg = threadIdx.x >> 3, kt = threadIdx.x & 7;
  float re = 0.0f, im = 0.0f;
  for (int kxi = 0; kxi < KXM; ++kxi) {
    int kx = (kxi < 12) ? kxi : kxi + 40;
    int j = (kx * xg) & 63;
    float gr = lr[kxi * 8 + kt], gi = li[kxi * 8 + kt];
    re += gr * c64[j] - gi * s64[j];
    im += gr * s64[j] + gi * c64[j];
  }
  long o = ((long)(bc * 64 + xg) * 64 + y) * MTM + kt;
  g2re[o] = re; g2im[o] = im;
}

// ---------------------------------------------------------------------------
// Inverse T (irfft, kt=0..7 nonzero) fused with +affine (RMW) and GELU:
//   io[t] += (1/81920)*( Re(H0) + 2*sum_{k=1..7}(hr*cos - hi*sin) ); gelu?
// ---------------------------------------------------------------------------
__global__ void k_combine(const float* __restrict__ g2re,
                          const float* __restrict__ g2im,
                          float* __restrict__ io, int doGelu) {
  __shared__ float ct[MTM * TO], st[MTM * TO];
  if (threadIdx.x < MTM * TO) {
    int k = threadIdx.x / TO, t = threadIdx.x % TO;
    float ang = 2.0f * PI_F * (float)(k * t) / (float)TO;
    float s, c; sincosf(ang, &s, &c);
    ct[threadIdx.x] = c; st[threadIdx.x] = s;
  }
  __syncthreads();
  int g = blockIdx.x * 256 + threadIdx.x;   // 1,048,576 exact
  const float norm = 1.0f / (64.0f * 64.0f * 20.0f);
  float hr[MTM], hi[MTM];
  long base = (long)g * MTM;
#pragma unroll
  for (int k = 0; k < MTM; ++k) { hr[k] = g2re[base + k]; hi[k] = g2im[base + k]; }
  float* op = io + (long)g * TO;
  for (int t = 0; t < TO; ++t) {
    float s = hr[0];
#pragma unroll
    for (int k = 1; k < MTM; ++k)
      s += 2.0f * (hr[k] * ct[k * TO + t] - hi[k] * st[k * TO + t]);
    float v = op[t] + norm * s;
    if (doGelu) v = gelu_f(v);
    op[t] = v;
  }
}

// ---------------------------------------------------------------------------
// Fused fc2 (32->128 WMMA + GELU) and fc3 (128->1 dot, shfl reduction).
// ---------------------------------------------------------------------------
__global__ void k_fc2fc3(const float* __restrict__ in,
                         const float* __restrict__ W2,
                         const float* __restrict__ b2,
                         const float* __restrict__ W3,
                         const float* __restrict__ b3,
                         float* __restrict__ out) {
  __shared__ float sW[CH * FCH];   // 32*128
  __shared__ float sB[FCH];
  __shared__ float s3[FCH];
  for (int i = threadIdx.x; i < CH * FCH; i += blockDim.x) sW[i] = W2[i];
  for (int i = threadIdx.x; i < FCH; i += blockDim.x) { sB[i] = b2[i]; s3[i] = W3[i]; }
  __syncthreads();

  int wave = threadIdx.x >> 5, lane = threadIdx.x & 31;
  int tile = blockIdx.x * 8 + wave;          // 40960 tiles exact
  int b = tile / (NPT >> 4);
  int p0 = (tile - b * (NPT >> 4)) << 4;
  const float* inb = in + (long)b * CH * NPT;
  int M = lane & 15;
  int khalf = (lane < 16) ? 0 : 2;

  // Preload all A fragments (K=32 -> 8 steps)
  v2f av[8];
#pragma unroll
  for (int ks = 0; ks < 8; ++ks) {
    int k = ks * 4 + khalf;
    av[ks][0] = inb[(long)k * NPT + p0 + M];
    av[ks][1] = inb[(long)(k + 1) * NPT + p0 + M];
    __builtin_prefetch(&inb[(long)k * NPT + p0 + M + 128], 0, 1);
  }

  float part[8];
#pragma unroll
  for (int r = 0; r < 8; ++r) part[r] = 0.0f;

#pragma unroll
  for (int f = 0; f < FCH / 16; ++f) {
    int co0 = f << 4;
    v8f acc = {};
#pragma unroll
    for (int ks = 0; ks < 8; ++ks) {
      int k = ks * 4 + khalf;
      v2f bb; bb[0] = sW[k * FCH + co0 + M];
              bb[1] = sW[(k + 1) * FCH + co0 + M];
      acc = wmma_f32(av[ks], bb, acc);
    }
    float bv = sB[co0 + M], w3 = s3[co0 + M];
#pragma unroll
    for (int r = 0; r < 8; ++r) part[r] += gelu_f(acc[r] + bv) * w3;
  }
  // Reduce across 16-lane half-groups (lanes 0-15: rows 0-7; 16-31: rows 8-15)
#pragma unroll
  for (int r = 0; r < 8; ++r) {
    part[r] += __shfl_xor(part[r], 1, 32);
    part[r] += __shfl_xor(part[r], 2, 32);
    part[r] += __shfl_xor(part[r], 4, 32);
    part[r] += __shfl_xor(part[r], 8, 32);
  }
  float bias3 = b3[0];
  float* op = out + (long)b * NPT + p0;
  if (lane == 0) {
#pragma unroll
    for (int r = 0; r < 8; ++r) op[r] = part[r] + bias3;
  } else if (lane == 16) {
#pragma unroll
    for (int r = 0; r < 8; ++r) op[8 + r] = part[r] + bias3;
  }
}

// ---------------------------------------------------------------------------
// Host-side launch chain (graph-capture safe: stream-only, no alloc/sync).
// ---------------------------------------------------------------------------
extern "C" void kernel_launch(void* const* d_in, const int* in_sizes, int n_in,
                              void* d_out, int out_size, void* d_ws, size_t ws_size,
                              hipStream_t stream) {
  const float* x     = (const float*)d_in[0];
  const float* Wfc0  = (const float*)d_in[1];
  const float* bfc0  = (const float*)d_in[2];
  const float* Wfc1  = (const float*)d_in[3];
  const float* bfc1  = (const float*)d_in[4];
  const float* Wfc2  = (const float*)d_in[5];
  const float* bfc2  = (const float*)d_in[6];
  const float* Wfc3  = (const float*)d_in[7];
  const float* bfc3  = (const float*)d_in[8];
  const float* Waff  = (const float*)d_in[9];
  const float* baff  = (const float*)d_in[10];
  const float* swre  = (const float*)d_in[11];
  const float* swim  = (const float*)d_in[12];
  float* outp = (float*)d_out;

  float* ws = (float*)d_ws;
  size_t o = 0;
  float* xA    = ws + o; o += (size_t)NB * CH * NPT;        // 20,971,520
  float* xB    = ws + o; o += (size_t)NB * CH * NPT;        // 20,971,520
  float* FtRe  = ws + o; o += (size_t)NB * CH * NYD * MTM * NXD; // 8,388,608
  float* FtIm  = ws + o; o += (size_t)NB * CH * NYD * MTM * NXD;
  float* FxRe  = ws + o; o += (size_t)NB * CH * KXM * MTM * NYD; // 3,145,728
  float* FxIm  = ws + o; o += (size_t)NB * CH * KXM * MTM * NYD;
  float* FxyRe = ws + o; o += (size_t)MODEN * CH * NB;           // 1,179,648
  float* FxyIm = ws + o; o += (size_t)MODEN * CH * NB;
  float* GRe   = ws + o; o += (size_t)NB * CH * MODEN;
  float* GIm   = ws + o; o += (size_t)NB * CH * MODEN;
  float* WrRe  = ws + o; o += (size_t)4 * 4 * MSZ * 1024;        // 18,874,368
  float* WrIm  = ws + o; o += (size_t)4 * 4 * MSZ * 1024;
  // Aliases (lifetimes are disjoint within each FNO block):
  float* x0   = FtRe;   // fc0 output (10.5M floats) lives before any Ft use
  float* GyRe = FxRe, *GyIm = FxIm;   // Fx consumed by yfwd before yinv writes
  float* G2Re = FtRe, *G2Im = FtIm;   // Ft consumed by xfwd before xinv writes

  // 1. Repack spectral weights to mode-major (once per call; ~150MB, trivial)
  k_repack<<<73728, 256, 0, stream>>>(swre, swim, WrRe, WrIm);

  // 2. Channel lift 4->32 + GELU  (x[8,4,40960] -> x0[8,32,40960])
  k_chgemm<4, CH, 1><<<2560, 256, 0, stream>>>(x, Wfc0, bfc0, x0, NP0);

  // 3. Time lift 10->20 + GELU  (x0 -> xA[8,32,81920])
  k_timelift<<<8192, 256, 0, stream>>>(x0, Wfc1, bfc1, xA);

  // 4. FNO blocks
  float* cur = xA;
  float* nxt = xB;
  for (int blk = 0; blk < 4; ++blk) {
    // parallel affine path -> nxt
    k_chgemm<CH, CH, 0><<<5120, 256, 0, stream>>>(cur, Waff + blk * CH * CH,
                                                  baff + blk * CH, nxt, NPT);
    // spectral path
    k_tfwd<<<4096, 256, 0, stream>>>(cur, FtRe, FtIm);
    k_xfwd<<<16384, 192, 0, stream>>>(FtRe, FtIm, FxRe, FxIm);
    k_yfwd<<<6144, 192, 0, stream>>>(FxRe, FxIm, FxyRe, FxyIm);
    k_mix<<<576, 256, 0, stream>>>(FxyRe, FxyIm, WrRe, WrIm, GRe, GIm, blk);
    k_yinv<<<6144, 512, 0, stream>>>(GRe, GIm, GyRe, GyIm);
    k_xinv<<<16384, 512, 0, stream>>>(GyRe, GyIm, G2Re, G2Im);
    // nxt += irfft(spectral); gelu except last block
    k_combine<<<4096, 256, 0, stream>>>(G2Re, G2Im, nxt, (blk < 3) ? 1 : 0);
    float* tmp = cur; cur = nxt; nxt = tmp;
  }

  // 5. Fused fc2(32->128,gelu) + fc3(128->1)
  k_fc2fc3<<<5120, 256, 0, stream>>>(cur, Wfc2, bfc2, Wfc3, bfc3, outp);
}